// CapsuleLayer_10033043603466
// MI455X (gfx1250) — compile-verified
//
#include <hip/hip_runtime.h>

// CapsuleLayer dynamic routing for MI455X (gfx1250, wave32), Gram-matrix form.
//
// u_hat[b,j,i,d] = sum_p inputs[b,i,p] * W[j,i,d,p]
// c_r[b,i] = 1 + odsum_r[b,:] . u_hat[b,i,:]   (odsum = sum of normalized outs)
// => v_r[b,d] = S[b,d] + sum_d' odsum_r[b,d'] * G[b,d',d]
//    S[b,d]    = sum_i u_hat[b,i,d]                 (WMMA, K=p)
//    G[b,d,d'] = sum_i u_hat[b,i,d]*u_hat[b,i,d']   (WMMA, K=i, A==B operand)
// One streaming pass over W builds S and G; routing is then a tiny matvec.

typedef __attribute__((ext_vector_type(2))) float v2f;
typedef __attribute__((ext_vector_type(8))) float v8f;

#define B_TOT 64
#define J_TOT 64
#define I_TOT 4096
#define D_DIM 16
#define P_DIM 8
#define WAVES 8
#define ROUTS 3

#define CH    32                // i per pipelined chunk
#define NCH   (I_TOT / CH)      // 128 chunks
#define SB    36                // float stride between b-slabs in a pair-row
#define PITCH 608               // floats per i-pair row (16*36 + 32 pad)
#define UBUF  ((CH / 2) * PITCH)

// dynamic LDS partition (in floats)
#define RED_OFF (2 * UBUF)              // S cross-wave staging (8 waves * 256)
#define S_OFF   (RED_OFF + WAVES * 256) // S[b][d]  (256)
#define G_OFF   (S_OFF + 256)           // G[b][d][d'] (4096)
#define LDS_FLOATS (G_OFF + 4096)

__global__ __launch_bounds__(256) void capsule_gram_wmma(
    const float* __restrict__ inp,   // [B, I, P]
    const float* __restrict__ Wm,    // [J, I, D, P]
    float* __restrict__ out)         // [B, J, D]
{
    extern __shared__ float sm[];

    const int tid  = threadIdx.x;
    const int lane = tid & 31;
    const int wave = tid >> 5;
    const int half = lane >> 4;     // WMMA K-half selector
    const int lm   = lane & 15;     // M/N index within a half

    const int j  = blockIdx.x >> 2;          // adjacent blocks share W[j]
    const int b0 = (blockIdx.x & 3) * 16;

    // u_hat producer operands (WMMA tile M=b, N=d, K=p):
    //   A: inputs[(b0+lm), i, 2*half + {0,1}]   B: W[j, i, lm, 2*half + {0,1}]
    const float* pa = inp + (size_t)(b0 + lm) * (I_TOT * P_DIM) + 2 * half
                          + (size_t)(wave * 4) * P_DIM;
    const float* pb = Wm + (size_t)j * ((size_t)I_TOT * D_DIM * P_DIM)
                         + lm * P_DIM + 2 * half
                         + (size_t)(wave * 4) * (D_DIM * P_DIM);

    v8f Sacc = {};          // sum_i u_hat, WMMA C/D layout
    v8f G0 = {}, G1 = {};   // Gram chains for b = wave and b = wave + 8
    const int blo = wave;
    const int bhi = wave + 8;

    // produce: this wave's 4 i of the next chunk -> u_hat into LDS buffer `buf`
    auto produce = [&](int buf) {
        float* ub = sm + buf * UBUF;
        v8f uprev = {};
        #pragma unroll
        for (int k = 0; k < 4; ++k) {
            v2f a0  = *(const v2f*)(pa);
            v2f a1  = *(const v2f*)(pa + 4);
            v2f bq0 = *(const v2f*)(pb);
            v2f bq1 = *(const v2f*)(pb + 4);
            v8f z = {};
            v8f u = __builtin_amdgcn_wmma_f32_16x16x4_f32(
                        false, a0, false, bq0, (short)0, z, false, false);
            u = __builtin_amdgcn_wmma_f32_16x16x4_f32(
                        false, a1, false, bq1, (short)0, u, false, false);
            Sacc += u;
            pa += P_DIM;
            pb += D_DIM * P_DIM;
            __builtin_prefetch(pb + (CH - 4) * D_DIM * P_DIM, 0, 1);
            if (k & 1) {
                // store the (i_even, i_odd) pair: b64 per accumulator register
                const int pairIdx = wave * 2 + (k >> 1);
                float* row = ub + pairIdx * PITCH + lm * 2;
                #pragma unroll
                for (int q = 0; q < 8; ++q) {
                    v2f pr;
                    pr[0] = uprev[q];   // i even
                    pr[1] = u[q];       // i odd
                    *(v2f*)(row + (q + 8 * half) * SB) = pr;
                }
            } else {
                uprev = u;
            }
        }
        pa += (CH - 4) * P_DIM;               // jump to next chunk
        pb += (size_t)(CH - 4) * D_DIM * P_DIM;
    };

    // consume: Gram update over one staged chunk (K = 32, 8 WMMAs per chain).
    // A and B operands of U^T @ U are the SAME lane values -> one b64 load each.
    auto consume = [&](int c) {
        const float* ub = sm + (c & 1) * UBUF + lm * 2;
        #pragma unroll
        for (int t = 0; t < 8; ++t) {         // i-quad i0 = 4t
            const float* src = ub + (2 * t + half) * PITCH;
            v2f ulo = *(const v2f*)(src + blo * SB);
            v2f uhi = *(const v2f*)(src + bhi * SB);
            G0 = __builtin_amdgcn_wmma_f32_16x16x4_f32(
                     false, ulo, false, ulo, (short)0, G0, false, false);
            G1 = __builtin_amdgcn_wmma_f32_16x16x4_f32(
                     false, uhi, false, uhi, (short)0, G1, false, false);
        }
    };

    // ---- single streaming pass over W: software-pipelined produce/consume ----
    produce(0);
    __syncthreads();
    for (int c = 0; c < NCH; ++c) {
        if (c + 1 < NCH) produce((c + 1) & 1);
        consume(c);
        __syncthreads();
    }

    // ---- stage S partials and Gram tiles into LDS ----
    #pragma unroll
    for (int q = 0; q < 8; ++q)
        sm[RED_OFF + wave * 256 + q * 32 + lane] = Sacc[q];
    #pragma unroll
    for (int q = 0; q < 8; ++q) {
        sm[G_OFF + (blo * 16 + q + 8 * half) * 16 + lm] = G0[q];
        sm[G_OFF + (bhi * 16 + q + 8 * half) * 16 + lm] = G1[q];
    }
    __syncthreads();

    {   // reduce S across the 8 waves: thread t owns one (reg, lane) slot
        float v = 0.0f;
        #pragma unroll
        for (int w = 0; w < WAVES; ++w) v += sm[RED_OFF + w * 256 + tid];
        const int bb = (tid >> 5) + 8 * ((tid >> 4) & 1);
        const int dd = tid & 15;
        sm[S_OFF + bb * 16 + dd] = v;
    }
    __syncthreads();

    // ---- routing: 3 rounds of 16x16 matvec per b (trivial work) ----
    if (tid < 16) {
        const int b = tid;
        float od[16];
        #pragma unroll
        for (int d = 0; d < 16; ++d) od[d] = 0.0f;

        for (int r = 0; r < ROUTS; ++r) {
            float vv[16];
            float n2 = 0.0f;
            #pragma unroll
            for (int d = 0; d < 16; ++d) {
                float x = sm[S_OFF + b * 16 + d];
                #pragma unroll
                for (int dp = 0; dp < 16; ++dp)
                    x += od[dp] * sm[G_OFF + (b * 16 + dp) * 16 + d];
                vv[d] = x;
                n2 += x * x;
            }
            const float inv = 1.0f / n2;   // reference: v / sum(v*v)
            if (r == ROUTS - 1) {
                #pragma unroll
                for (int d = 0; d < 16; ++d)
                    out[((size_t)(b0 + b) * J_TOT + j) * D_DIM + d] = vv[d] * inv;
            } else {
                #pragma unroll
                for (int d = 0; d < 16; ++d) od[d] += vv[d] * inv;
            }
        }
    }
}

extern "C" void kernel_launch(void* const* d_in, const int* in_sizes, int n_in,
                              void* d_out, int out_size, void* d_ws, size_t ws_size,
                              hipStream_t stream) {
    const float* inp = (const float*)d_in[0];   // [64, 4096, 8]
    const float* Wm  = (const float*)d_in[1];   // [64, 4096, 16, 8]
    float* out = (float*)d_out;                 // [64, 64, 16]
    (void)in_sizes; (void)n_in; (void)out_size; (void)d_ws; (void)ws_size;

    dim3 grid(J_TOT * (B_TOT / 16));            // 256 blocks: (j, b-tile)
    dim3 block(256);                            // 8 wave32 waves
    size_t shmem = (size_t)LDS_FLOATS * sizeof(float);   // ~101 KB < 320 KB/WGP
    capsule_gram_wmma<<<grid, block, shmem, stream>>>(inp, Wm, out);
}